// momentumModule_47021301957200
// MI455X (gfx1250) — compile-verified
//
#include <hip/hip_runtime.h>

#ifndef M_PI
#define M_PI 3.14159265358979323846
#endif

// ext_vector types: valid operands for __builtin_nontemporal_load (HIP's
// int4/float4 are structs and are not).
typedef int   v4i __attribute__((ext_vector_type(4)));
typedef float v4f __attribute__((ext_vector_type(4)));

// ---------------------------------------------------------------------------
// Main edge kernel (non-template, defined FIRST so the disasm snippet shows
// its body): 4 edges per thread, grid-stride, non-temporal b128 streaming
// loads (th:TH_LOAD_NT -> read-once lines evict first, protecting the
// L2-resident particle/output working set), gfx1250 global_prefetch_b8 one
// grid-stride ahead, and non-returning global_atomic_add_f32 scatter.
// `fused` is a wave-uniform flag -> scalar branch, ~free.
// ---------------------------------------------------------------------------
__global__ __launch_bounds__(256) void momentum_edges_vec4(
    const v4i*    __restrict__ iv4,   // [E/4]
    const v4i*    __restrict__ jv4,   // [E/4]
    const v4f*    __restrict__ d4,    // [E/2]: 2x v4f per chunk of 4 edges
    const v4f*    __restrict__ q4,    // [E/4]
    const float*  __restrict__ Vj,    // [N]  (used when !fused)
    const float*  __restrict__ rhoj,  // [N]  (used when !fused)
    const float2* __restrict__ ui2,   // [N]
    const float2* __restrict__ uj2,   // [N]
    const float*  __restrict__ vrho,  // [N]  (used when fused)
    float*        __restrict__ out,   // [N]
    int nChunks, float coef, int fused)
{
    const int stride = gridDim.x * blockDim.x;
    for (int c = blockIdx.x * blockDim.x + threadIdx.x; c < nChunks; c += stride) {
        // Software prefetch of next grid-stride chunk (global_prefetch_b8).
        const int cn = c + stride;
        if (cn < nChunks) {
            __builtin_prefetch(iv4 + cn, 0, 3);
            __builtin_prefetch(jv4 + cn, 0, 3);
            __builtin_prefetch(q4  + cn, 0, 3);
            __builtin_prefetch(d4 + 2 * cn,     0, 3);
            __builtin_prefetch(d4 + 2 * cn + 1, 0, 3);
        }

        // Streaming reads: read-once -> non-temporal b128 loads.
        const v4i iv = __builtin_nontemporal_load(iv4 + c);
        const v4i jv = __builtin_nontemporal_load(jv4 + c);
        const v4f qv = __builtin_nontemporal_load(q4 + c);
        const v4f dA = __builtin_nontemporal_load(d4 + 2 * c);      // e0,e1 dirs
        const v4f dB = __builtin_nontemporal_load(d4 + 2 * c + 1);  // e2,e3 dirs

        const int   is[4]  = {iv.x, iv.y, iv.z, iv.w};
        const int   js[4]  = {jv.x, jv.y, jv.z, jv.w};
        const float qs[4]  = {qv.x, qv.y, qv.z, qv.w};
        const float dxs[4] = {dA.x, dA.z, dB.x, dB.z};
        const float dys[4] = {dA.y, dA.w, dB.y, dB.w};

#pragma unroll
        for (int k = 0; k < 4; ++k) {
            const int ii = is[k];
            const int jj = js[k];
            const float wj  = fused ? vrho[jj] : (Vj[jj] * rhoj[jj]);  // L2-resident gather
            const float omq = 1.0f - qs[k];
            const float dwdq = coef * qs[k] * omq * omq * omq;          // Wendland C2 dW/dq
            const float2 a = ui2[ii];                                   // b64 gather (RT hint)
            const float2 b = uj2[jj];                                   // b64 gather (RT hint)
            const float prod = (b.x - a.x) * dxs[k] + (b.y - a.y) * dys[k];
            // out[i] -= prod * dwdq * wj : HW non-returning global_atomic_add_f32
            unsafeAtomicAdd(out + ii, -(prod * dwdq * wj));
        }
    }
}

// ---------------------------------------------------------------------------
// Scalar tail for E % 4 leftover edges (<= 3)
// ---------------------------------------------------------------------------
__global__ void momentum_edges_tail(
    const int* __restrict__ i, const int* __restrict__ j,
    const float* __restrict__ d, const float* __restrict__ q,
    const float* __restrict__ Vj, const float* __restrict__ rhoj,
    const float2* __restrict__ ui2, const float2* __restrict__ uj2,
    const float* __restrict__ vrho, int fused,
    float* __restrict__ out, int start, int E, float coef)
{
    int e = start + (int)threadIdx.x;
    if (e >= E) return;
    const int ii = i[e];
    const int jj = j[e];
    const float qe = q[e];
    const float dx = d[2 * e], dy = d[2 * e + 1];
    const float wj = fused ? vrho[jj] : (Vj[jj] * rhoj[jj]);
    const float omq = 1.0f - qe;
    const float dwdq = coef * qe * omq * omq * omq;
    const float2 a = ui2[ii];
    const float2 b = uj2[jj];
    const float prod = (b.x - a.x) * dx + (b.y - a.y) * dy;
    unsafeAtomicAdd(out + ii, -(prod * dwdq * wj));
}

// ---------------------------------------------------------------------------
// Fused init: out[n] = 0 (harness poisons d_out) and vrho[n] = Vj[n]*rhoj[n]
// ---------------------------------------------------------------------------
__global__ __launch_bounds__(256) void init_kernel(const float* __restrict__ Vj,
                                                   const float* __restrict__ rhoj,
                                                   float* __restrict__ vrho,
                                                   float* __restrict__ out,
                                                   int n, int fused) {
    int t = blockIdx.x * blockDim.x + threadIdx.x;
    if (t < n) {
        out[t] = 0.0f;
        if (fused) vrho[t] = Vj[t] * rhoj[t];
    }
}

extern "C" void kernel_launch(void* const* d_in, const int* in_sizes, int n_in,
                              void* d_out, int out_size, void* d_ws, size_t ws_size,
                              hipStream_t stream) {
    // setup_inputs() order: i, j, distances, radialDistances, Vj, rhoj, ui, uj, numParticles
    const int*   i    = (const int*)d_in[0];
    const int*   j    = (const int*)d_in[1];
    const float* d    = (const float*)d_in[2];
    const float* q    = (const float*)d_in[3];
    const float* Vj   = (const float*)d_in[4];
    const float* rhoj = (const float*)d_in[5];
    const float2* ui2 = (const float2*)d_in[6];
    const float2* uj2 = (const float2*)d_in[7];
    float* out = (float*)d_out;

    const int E = in_sizes[0];
    const int N = out_size;

    // Wendland C2 2D: dW/dq = -20 * (7/pi) / h^3 * q * (1-q)^3  (h = 0.05)
    const float coef = (float)(-20.0 * (7.0 / M_PI) / (0.05 * 0.05 * 0.05));

    // 1) zero the output + optional Vj*rhoj fusion into workspace
    const int fused = (ws_size >= (size_t)N * sizeof(float)) ? 1 : 0;
    float* vrho = (float*)d_ws;
    init_kernel<<<(N + 255) / 256, 256, 0, stream>>>(Vj, rhoj, vrho, out, N, fused);

    // 2) main edge sweep: 4 edges/thread, grid-stride so the software prefetch
    //    has a target one stride ahead (~3 iterations per thread at E=6.4M).
    const int nChunks = E / 4;
    if (nChunks > 0) {
        int blocks = (nChunks + 255) / 256;
        if (blocks > 2048) blocks = 2048;
        momentum_edges_vec4<<<blocks, 256, 0, stream>>>(
            (const v4i*)i, (const v4i*)j, (const v4f*)d, (const v4f*)q,
            Vj, rhoj, ui2, uj2, vrho, out, nChunks, coef, fused);
    }

    // 3) tail (E % 4 edges)
    const int rem = E - nChunks * 4;
    if (rem > 0) {
        momentum_edges_tail<<<1, rem, 0, stream>>>(
            i, j, d, q, Vj, rhoj, ui2, uj2, vrho, fused,
            out, nChunks * 4, E, coef);
    }
}